// Net_7310034337831
// MI455X (gfx1250) — compile-verified
//
#include <hip/hip_runtime.h>

typedef float v2f __attribute__((ext_vector_type(2)));
typedef float v8f __attribute__((ext_vector_type(8)));

#define BATCH 16
#define SEQ   2048
#define HID   1024
#define NCLS  3
#define NWORD 512

// ---------------------------------------------------------------------------
// Kernel 1: token-level projection  tok[t, 0..2] = x[t, :] @ W  (K = 1024)
// One wave32 per 16-token tile, V_WMMA_F32_16X16X4_F32 accumulation.
// B fragments pre-packed in LDS as K-pairs so each step is one ds_load_b64
// with a constant immediate offset.
// ---------------------------------------------------------------------------
__global__ __launch_bounds__(256) void tok_project_wmma(
    const float* __restrict__ x,    // [B*S, H] row-major
    const float* __restrict__ W,    // [H, C] row-major
    float* __restrict__ tok)        // [B*S, C] workspace
{
    // Wp[q][n] = { W[2q][n], W[2q+1][n] }  for n<3, else {0,0}.  64 KB.
    __shared__ v2f Wp[(HID / 2) * 16];
    const int tid = threadIdx.x;
    for (int i = tid; i < (HID / 2) * 16; i += 256) {
        const int q = i >> 4;
        const int n = i & 15;
        v2f p;
        p.x = (n < NCLS) ? W[(2 * q + 0) * NCLS + n] : 0.0f;
        p.y = (n < NCLS) ? W[(2 * q + 1) * NCLS + n] : 0.0f;
        Wp[i] = p;
    }
    __syncthreads();

    const int wave = tid >> 5;
    const int lane = tid & 31;
    const int half = lane >> 4;   // 0: K pair {k,k+1}, 1: K pair {k+2,k+3}
    const int m    = lane & 15;   // A: row / B,D: column

    const int  tile = blockIdx.x * 8 + wave;        // 256 blocks * 8 waves = 2048 tiles
    const long row  = (long)tile * 16 + m;          // token index for A rows
    const float* xrow   = x + row * HID;
    const v2f*   wpLane = Wp + half * 16 + m;       // q = k/2 + half -> wpLane[(k/2)*16]

    v8f acc = {};
    for (int kk = 0; kk < HID; kk += 32) {
        // near-scope speculative prefetch 1KB ahead in this lane's row
        __builtin_prefetch(xrow + kk + 256, 0, 3);
        const float* xp = xrow + kk + 2 * half;     // A base for this kk block
        const v2f*   wp = wpLane + (kk >> 1) * 16;  // B base for this kk block
#pragma unroll
        for (int k4 = 0; k4 < 32; k4 += 4) {
            // A fragment: lanes 0-15 hold K=k,k+1 ; lanes 16-31 hold K=k+2,k+3
            v2f a = *(const v2f*)(xp + k4);
            // B fragment: matching K-pair, column n = m; immediate LDS offset
            v2f b = wp[(k4 >> 1) * 16];
            acc = __builtin_amdgcn_wmma_f32_16x16x4_f32(
                /*neg_a=*/false, a, /*neg_b=*/false, b,
                /*c_mod=*/(short)0, acc, /*reuse_a=*/false, /*reuse_b=*/false);
        }
    }

    // D layout (16x16 f32): VGPR v, lanes 0-15 -> M=v,  N=lane
    //                               lanes 16-31 -> M=v+8, N=lane-16
    if (m < NCLS) {
#pragma unroll
        for (int v = 0; v < 8; ++v) {
            const int mrow = v + 8 * half;
            tok[((long)tile * 16 + mrow) * NCLS + m] = acc[v];
        }
    }
}

// ---------------------------------------------------------------------------
// Kernel 2: per-sample segment mean of the 3-vector token logits, + bias,
// then gather back to token positions. One block per batch sample.
// ---------------------------------------------------------------------------
__global__ __launch_bounds__(256) void pool_gather(
    const int*   __restrict__ wid,   // [B*S]
    const float* __restrict__ tok,   // [B*S, C]
    const float* __restrict__ bias,  // [C]
    float*       __restrict__ out)   // [B*S, C]
{
    __shared__ float wsum[NWORD * 3];
    __shared__ float wcnt[NWORD];
    const int b   = blockIdx.x;
    const int tid = threadIdx.x;

    for (int i = tid; i < NWORD * 3; i += 256) wsum[i] = 0.0f;
    for (int i = tid; i < NWORD;     i += 256) wcnt[i] = 0.0f;
    __syncthreads();

    const int base = b * SEQ;
    for (int t = tid; t < SEQ; t += 256) {
        const int  w  = wid[base + t];
        const long ti = (long)(base + t) * NCLS;
        atomicAdd(&wsum[w * 3 + 0], tok[ti + 0]);   // ds_add_f32
        atomicAdd(&wsum[w * 3 + 1], tok[ti + 1]);
        atomicAdd(&wsum[w * 3 + 2], tok[ti + 2]);
        atomicAdd(&wcnt[w], 1.0f);
    }
    __syncthreads();

    const float b0 = bias[0], b1 = bias[1], b2 = bias[2];
    for (int w = tid; w < NWORD; w += 256) {
        const float inv = 1.0f / fmaxf(wcnt[w], 1.0f);
        wsum[w * 3 + 0] = wsum[w * 3 + 0] * inv + b0;
        wsum[w * 3 + 1] = wsum[w * 3 + 1] * inv + b1;
        wsum[w * 3 + 2] = wsum[w * 3 + 2] * inv + b2;
    }
    __syncthreads();

    for (int t = tid; t < SEQ; t += 256) {
        const int  w  = wid[base + t];
        const long oi = (long)(base + t) * NCLS;
        out[oi + 0] = wsum[w * 3 + 0];
        out[oi + 1] = wsum[w * 3 + 1];
        out[oi + 2] = wsum[w * 3 + 2];
    }
}

// ---------------------------------------------------------------------------
extern "C" void kernel_launch(void* const* d_in, const int* in_sizes, int n_in,
                              void* d_out, int out_size, void* d_ws, size_t ws_size,
                              hipStream_t stream) {
    const float* x    = (const float*)d_in[0];  // [16,2048,1024] f32
    const int*   wid  = (const int*)  d_in[1];  // [16,2048] i32 (sorted per row)
    const float* W    = (const float*)d_in[2];  // [1024,3] f32
    const float* bias = (const float*)d_in[3];  // [3] f32
    float* out = (float*)d_out;                 // [16,2048,3] f32
    float* tok = (float*)d_ws;                  // needs B*S*3*4 = 384 KiB scratch

    // 2048 tiles of 16 tokens, 8 waves (tiles) per 256-thread block.
    tok_project_wmma<<<256, 256, 0, stream>>>(x, W, tok);
    pool_gather<<<BATCH, 256, 0, stream>>>(wid, tok, bias, out);
}